// DetLoss_69715909149247
// MI455X (gfx1250) — compile-verified
//
#include <hip/hip_runtime.h>
#include <hip/hip_bf16.h>
#include <math.h>

#define A_TOT 82944     // 96*96*9 anchors
#define NGT   32
#define NCLS  80
#define NB    4
#define EMB   128
#define IDS   500
#define IDS_PAD 512
#define HW    9216      // 96*96
#define FSTR  132       // padded LDS row stride for feat (bank-conflict avoidance)

typedef __attribute__((ext_vector_type(2))) float v2f;
typedef __attribute__((ext_vector_type(8))) float v8f;

// acc layout (floats): [0..3]=num_pos, [4..7]=cls_sum, [8..11]=reg_sum, [12..15]=trk
__global__ void init_kernel(float* __restrict__ acc) {
    if (threadIdx.x < 16) acc[threadIdx.x] = 0.0f;
}

// ---------------- IoU assignment + smooth-L1 regression loss ----------------
__global__ void assign_kernel(const float* __restrict__ anchors,  // [A,4] (y1,x1,y2,x2)
                              const float* __restrict__ reg,      // [B,A,4]
                              const float* __restrict__ boxes,    // [B,32,4] (x1,y1,x2,y2)
                              const int*   __restrict__ labels,   // [B,32]
                              unsigned char* __restrict__ state,  // [B,A]: label | 128=neg | 255=ignore
                              float* __restrict__ acc) {
    const int b = blockIdx.y;
    const int tid = threadIdx.x;
    const int a = blockIdx.x * blockDim.x + tid;

    __shared__ float sb[NGT * 4];
    __shared__ int   slab[NGT];
    __shared__ float sredf[256];
    __shared__ float sredp[256];
    if (tid < NGT * 4) sb[tid] = boxes[b * NGT * 4 + tid];
    if (tid < NGT)     slab[tid] = labels[b * NGT + tid];
    __syncthreads();

    float regContrib = 0.0f;
    float isPos = 0.0f;
    if (a < A_TOT) {
        const float a0 = anchors[a * 4 + 0], a1 = anchors[a * 4 + 1];
        const float a2 = anchors[a * 4 + 2], a3 = anchors[a * 4 + 3];
        const float areaA = (a2 - a0) * (a3 - a1);
        float best = -1.0f; int arg = 0;
        for (int n = 0; n < NGT; ++n) {
            const float b0 = sb[n * 4 + 0], b1 = sb[n * 4 + 1];
            const float b2 = sb[n * 4 + 2], b3 = sb[n * 4 + 3];
            float iw = fmaxf(fminf(a3, b2) - fmaxf(a1, b0), 0.0f);
            float ih = fmaxf(fminf(a2, b3) - fmaxf(a0, b1), 0.0f);
            float inter = iw * ih;
            float ua = fmaxf(areaA + (b2 - b0) * (b3 - b1) - inter, 1e-8f);
            float iou = inter / ua;
            if (iou > best) { best = iou; arg = n; }   // first-max == jnp.argmax
        }
        const bool pos = best >= 0.5f;
        const bool neg = best < 0.4f;
        unsigned char st = 255;
        if (pos)      st = (unsigned char)slab[arg];
        else if (neg) st = 128;
        state[(size_t)b * A_TOT + a] = st;

        if (pos) {
            isPos = 1.0f;
            const float ab0 = sb[arg * 4 + 0], ab1 = sb[arg * 4 + 1];
            const float ab2 = sb[arg * 4 + 2], ab3 = sb[arg * 4 + 3];
            float gw = ab2 - ab0, gh = ab3 - ab1;
            const float gcx = ab0 + 0.5f * gw, gcy = ab1 + 0.5f * gh; // unclipped (per ref)
            gw = fmaxf(gw, 1.0f); gh = fmaxf(gh, 1.0f);
            const float aw = a3 - a1, ah = a2 - a0;
            const float acx = a1 + 0.5f * aw, acy = a0 + 0.5f * ah;
            float t[4];
            t[0] = (gcy - acy) / ah;
            t[1] = (gcx - acx) / aw;
            t[2] = __logf(gh / ah);
            t[3] = __logf(gw / aw);
            const float* rg = reg + ((size_t)b * A_TOT + a) * 4;
            #pragma unroll
            for (int j = 0; j < 4; ++j) {
                float d = fabsf(t[j] - rg[j]);
                regContrib += (d <= (1.0f / 9.0f)) ? 4.5f * d * d : d - (0.5f / 9.0f);
            }
        }
    }
    sredf[tid] = regContrib;
    sredp[tid] = isPos;
    __syncthreads();
    for (int s = 128; s > 0; s >>= 1) {
        if (tid < s) { sredf[tid] += sredf[tid + s]; sredp[tid] += sredp[tid + s]; }
        __syncthreads();
    }
    if (tid == 0) {
        if (sredp[0] != 0.0f) atomicAdd(&acc[0 + b], sredp[0]);
        if (sredf[0] != 0.0f) atomicAdd(&acc[8 + b], sredf[0]);
    }
}

// ---------------- focal classification loss (bandwidth-bound stream) ----------------
__global__ void focal_kernel(const float* __restrict__ cls,       // [B,A,80]
                             const unsigned char* __restrict__ state,
                             float* __restrict__ acc) {
    const int b = blockIdx.y;
    const int tid = threadIdx.x;
    const long long q = (long long)blockIdx.x * blockDim.x + tid;  // quad index: A*20 per image
    __shared__ float sred[256];

    float contrib = 0.0f;
    if (q < (long long)A_TOT * 20) {
        const int a    = (int)(q / 20);
        const int quad = (int)(q % 20);
        const unsigned char st = state[(size_t)b * A_TOT + a];
        if (st != 255) {  // skip the 106MB read for ignore anchors
            const float4 p4 = *(const float4*)(cls + ((size_t)b * A_TOT + a) * NCLS + quad * 4);
            const int lab = (st < 128) ? (int)st : -1;
            float pv[4] = {p4.x, p4.y, p4.z, p4.w};
            #pragma unroll
            for (int j = 0; j < 4; ++j) {
                float p = fminf(fmaxf(pv[j], 1e-4f), 1.0f - 1e-4f);
                const int c = quad * 4 + j;
                if (lab == c) contrib += 0.25f * (1.0f - p) * (1.0f - p) * (-__logf(p));
                else          contrib += 0.75f * p * p * (-__logf(1.0f - p));
            }
        }
    }
    sred[tid] = contrib;
    __syncthreads();
    for (int s = 128; s > 0; s >>= 1) {
        if (tid < s) sred[tid] += sred[tid + s];
        __syncthreads();
    }
    if (tid == 0 && sred[0] != 0.0f) atomicAdd(&acc[4 + b], sred[0]);
}

// ---------------- ReID: gather + normalize + 32x500x128 GEMM via V_WMMA_F32_16X16X4_F32 ----------------
__global__ void reid_gemm_kernel(const float* __restrict__ emb,     // [B,128,9216]
                                 const float* __restrict__ boxes,   // [B,32,4]
                                 const float* __restrict__ Wt,      // [128,500]
                                 const float* __restrict__ bt,      // [500]
                                 const float* __restrict__ emb_scale,
                                 float* __restrict__ logits) {      // [B,32,512]
    const int b = blockIdx.x;
    const int tid = threadIdx.x;          // 128 threads = 4 waves
    __shared__ float feat[NGT][FSTR];
    __shared__ int   sind[NGT];
    __shared__ float sscale[NGT];

    if (tid < NGT) {
        const float* bx = boxes + (b * NGT + tid) * 4;
        const int cx = (int)floorf((bx[0] + bx[2]) * 0.125f);
        const int cy = (int)floorf((bx[1] + bx[3]) * 0.125f);
        sind[tid] = cy * 96 + cx;
    }
    __syncthreads();
    for (int i = tid; i < NGT * EMB; i += 128) {
        const int n = i >> 7, c = i & 127;
        feat[n][c] = emb[((size_t)b * EMB + c) * HW + sind[n]];
    }
    __syncthreads();
    if (tid < NGT) {
        float ss = 0.0f;
        for (int c = 0; c < EMB; ++c) { const float v = feat[tid][c]; ss += v * v; }
        sscale[tid] = emb_scale[0] / fmaxf(sqrtf(ss), 1e-12f);
    }
    __syncthreads();
    for (int i = tid; i < NGT * EMB; i += 128) {
        const int n = i >> 7, c = i & 127;
        feat[n][c] *= sscale[n];
    }
    __syncthreads();

    const int wave = tid >> 5;
    const int lane = tid & 31;
    const int m    = lane & 15;        // A-operand row / D-operand col index
    const int half = lane >> 4;        // K/M half-select per ISA layout
    // 64 (mt,nt) tile pairs: mt in {0,1} (M=32), nt in 0..31 (N=512 pad); 16 pairs/wave
    for (int p = wave * 16; p < wave * 16 + 16; ++p) {
        const int mt  = p & 1;
        const int nt  = p >> 1;
        const int col = nt * 16 + (lane & 15);
        const int colc = col < IDS ? col : (IDS - 1);
        const float msk = (col < IDS) ? 1.0f : 0.0f;
        v8f c = {};
        for (int k = 0; k < EMB; k += 4) {
            v2f av, bv;
            av.x = feat[mt * 16 + m][k + 2 * half + 0];
            av.y = feat[mt * 16 + m][k + 2 * half + 1];
            bv.x = Wt[(k + 2 * half + 0) * IDS + colc] * msk;
            bv.y = Wt[(k + 2 * half + 1) * IDS + colc] * msk;
            c = __builtin_amdgcn_wmma_f32_16x16x4_f32(
                    false, av, false, bv, (short)0, c, false, false);
        }
        const float bias = bt[colc] * msk;
        float* outp = logits + (size_t)b * NGT * IDS_PAD;
        #pragma unroll
        for (int r = 0; r < 8; ++r) {
            const int row = mt * 16 + r + 8 * half;
            outp[row * IDS_PAD + col] = c[r] + bias;
        }
    }
}

// ---------------- ReID: log-softmax CE over 500 ids ----------------
__global__ void reid_loss_kernel(const float* __restrict__ logits,  // [B,32,512]
                                 const int*   __restrict__ reids,   // [B,32]
                                 const float* __restrict__ acc_in,  // num_pos at [0..3]
                                 float* __restrict__ acc) {
    const int b = blockIdx.x;
    const int tid = threadIdx.x;          // 256 threads = 8 waves
    const int wave = tid >> 5, lane = tid & 31;
    __shared__ float swsum[8];
    __shared__ float swcnt[8];

    float ceSum = 0.0f, vcnt = 0.0f;
    for (int row = wave; row < NGT; row += 8) {
        const float* lp = logits + ((size_t)b * NGT + row) * IDS_PAD;
        float mx = -1e30f;
        for (int c = lane; c < IDS; c += 32) mx = fmaxf(mx, lp[c]);
        for (int off = 16; off; off >>= 1) mx = fmaxf(mx, __shfl_xor(mx, off));
        float se = 0.0f;
        for (int c = lane; c < IDS; c += 32) se += __expf(lp[c] - mx);
        for (int off = 16; off; off >>= 1) se += __shfl_xor(se, off);
        if (lane == 0) {
            const int rid = reids[b * NGT + row];
            const float valid = (rid >= 0) ? 1.0f : 0.0f;
            const int tgt = (rid > 0) ? rid : 0;
            ceSum += -(lp[tgt] - mx - __logf(se)) * valid;
            vcnt += valid;
        }
    }
    if (lane == 0) { swsum[wave] = ceSum; swcnt[wave] = vcnt; }
    __syncthreads();
    if (tid == 0) {
        float s = 0.0f, c = 0.0f;
        for (int w = 0; w < 8; ++w) { s += swsum[w]; c += swcnt[w]; }
        const float idl = s / fmaxf(c, 1.0f);
        acc[12 + b] = (acc_in[b] > 0.0f) ? idl : 0.0f;
    }
}

__global__ void finalize_kernel(const float* __restrict__ acc, float* __restrict__ out) {
    if (threadIdx.x == 0) {
        float cls = 0.0f, reg = 0.0f, trk = 0.0f;
        for (int b = 0; b < NB; ++b) {
            const float np = acc[b];
            cls += acc[4 + b] / fmaxf(np, 1.0f);
            reg += acc[8 + b] / fmaxf(np * 4.0f, 1.0f);
            trk += acc[12 + b];
        }
        out[0] = cls * 0.25f;
        out[1] = reg * 0.25f * 50.0f;
        out[2] = trk * 0.25f;
    }
}

extern "C" void kernel_launch(void* const* d_in, const int* in_sizes, int n_in,
                              void* d_out, int out_size, void* d_ws, size_t ws_size,
                              hipStream_t stream) {
    const float* cls      = (const float*)d_in[0];   // [4,82944,80]
    const float* reg      = (const float*)d_in[1];   // [4,82944,4]
    const float* anchors  = (const float*)d_in[2];   // [1,82944,4]
    const float* emb      = (const float*)d_in[3];   // [4,128,96,96]
    const float* boxes    = (const float*)d_in[4];   // [4,32,4]
    const int*   labels   = (const int*)d_in[5];     // [4,32]
    const int*   reids    = (const int*)d_in[6];     // [4,32]
    const float* Wt       = (const float*)d_in[7];   // [128,500]
    const float* bt       = (const float*)d_in[8];   // [500]
    const float* escale   = (const float*)d_in[9];   // [1]
    float* out = (float*)d_out;

    // workspace layout
    unsigned char* ws_state = (unsigned char*)d_ws;                        // 4*82944 B
    float* ws_acc    = (float*)((char*)d_ws + 4 * A_TOT);                  // 16 floats
    float* ws_logits = (float*)((char*)d_ws + 4 * A_TOT + 256);           // 4*32*512 floats

    init_kernel<<<1, 32, 0, stream>>>(ws_acc);

    dim3 agrid((A_TOT + 255) / 256, NB);
    assign_kernel<<<agrid, 256, 0, stream>>>(anchors, reg, boxes, labels, ws_state, ws_acc);

    dim3 fgrid((A_TOT * 20 + 255) / 256, NB);
    focal_kernel<<<fgrid, 256, 0, stream>>>(cls, ws_state, ws_acc);

    reid_gemm_kernel<<<NB, 128, 0, stream>>>(emb, boxes, Wt, bt, escale, ws_logits);
    reid_loss_kernel<<<NB, 256, 0, stream>>>(ws_logits, reids, ws_acc, ws_acc);

    finalize_kernel<<<1, 32, 0, stream>>>(ws_acc, out);
}